// DynamicPillarFeatureNet_17454747091077
// MI455X (gfx1250) — compile-verified
//
#include <hip/hip_runtime.h>

#define GRID_ 512
#define NV_ (GRID_ * GRID_)
#define BATCH_ 2
#define COUT_ 64
#define NSEG_ (BATCH_ * NV_ + 1)

typedef float v2f __attribute__((ext_vector_type(2)));
typedef float v8f __attribute__((ext_vector_type(8)));

// ---------------- zero-fill ----------------
__global__ void zero_f(float* __restrict__ p, long nelem) {
    long i = (long)blockIdx.x * blockDim.x + threadIdx.x;
    long stride = (long)gridDim.x * blockDim.x;
    for (; i < nelem; i += stride) p[i] = 0.0f;
}

// ---------------- pass 1: per-pillar count & sum(xyz) ----------------
__global__ void pillar_stats(const float* __restrict__ pts, float* __restrict__ ws, int n) {
    int i = blockIdx.x * blockDim.x + threadIdx.x;
    if (i >= n) return;
    const float* pt = pts + (size_t)i * 5;
    float bi = pt[0], x = pt[1], y = pt[2], z = pt[3];
    float cfx = (x - (-51.2f)) / 0.2f;
    float cfy = (y - (-51.2f)) / 0.2f;
    bool m = (cfx >= 0.f) && (cfx < 512.f) && (cfy >= 0.f) && (cfy < 512.f);
    int cx = (int)floorf(cfx);
    int cy = (int)floorf(cfy);
    int b = (int)bi;
    int seg = m ? (b * NV_ + cy * GRID_ + cx) : (BATCH_ * NV_);
    float* cnt = ws;
    float* sx  = ws + (size_t)NSEG_;
    float* sy  = ws + (size_t)2 * NSEG_;
    float* sz  = ws + (size_t)3 * NSEG_;
    atomicAdd(&cnt[seg], 1.0f);
    atomicAdd(&sx[seg], x);
    atomicAdd(&sy[seg], y);
    atomicAdd(&sz[seg], z);
}

// ---------------- pass 2: features -> WMMA GEMM -> BN/ReLU -> atomicMax scatter ----------------
// Each wave handles groups of 16 points. A is 16x12 (K padded from 9), B is 12x64 (W),
// done as 4 N-tiles x 3 chained V_WMMA_F32_16X16X4_F32.
__global__ void __launch_bounds__(256)
pillar_wmma(const float* __restrict__ pts,
            const float* __restrict__ Wm,
            const float* __restrict__ gamma,
            const float* __restrict__ beta,
            const float* __restrict__ rmean,
            const float* __restrict__ rvar,
            const float* __restrict__ ws,
            float* __restrict__ out,
            int n, int ngroups) {
    const int lane = threadIdx.x & 31;
    const int col  = lane & 15;   // N (channel-within-tile) for B/C frags; point row for A frag data
    const int half = lane >> 4;   // K-half selector
    const int wave   = (blockIdx.x * blockDim.x + threadIdx.x) >> 5;
    const int nwaves = (gridDim.x * blockDim.x) >> 5;

    // ---- Preload B fragments (W is 9x64 row-major, zero-padded to K=12) ----
    // frag[t] covers k = 4t + 2*half + {0,1}, column = 16*nn + col
    v2f bf[4][3];
#pragma unroll
    for (int t = 0; t < 3; ++t) {
        int k0 = 4 * t + 2 * half;
        int k1 = k0 + 1;
        int k0c = k0 < 8 ? k0 : 8;   // clamp to stay in-bounds, zero after load
        int k1c = k1 < 8 ? k1 : 8;
#pragma unroll
        for (int nn = 0; nn < 4; ++nn) {
            int c = 16 * nn + col;
            float b0 = Wm[k0c * COUT_ + c];
            float b1 = Wm[k1c * COUT_ + c];
            if (k0 > 8) b0 = 0.f;
            if (k1 > 8) b1 = 0.f;
            bf[nn][t] = (v2f){b0, b1};
        }
    }

    // ---- Fold BN into scale/bias per channel ----
    float scl[4], bia[4];
#pragma unroll
    for (int nn = 0; nn < 4; ++nn) {
        int c = 16 * nn + col;
        float inv = rsqrtf(rvar[c] + 1.0e-3f);
        float s = gamma[c] * inv;
        scl[nn] = s;
        bia[nn] = beta[c] - rmean[c] * s;
    }

    const float* cnt = ws;
    const float* sxp = ws + (size_t)NSEG_;
    const float* syp = ws + (size_t)2 * NSEG_;
    const float* szp = ws + (size_t)3 * NSEG_;

    for (int g = wave; g < ngroups; g += nwaves) {
        int p = g * 16 + col;
        bool inb = p < n;
        int pc = inb ? p : (n - 1);
        const float* pt = pts + (size_t)pc * 5;
        float bi = pt[0], x = pt[1], y = pt[2], z = pt[3], r = pt[4];

        float cfx = (x - (-51.2f)) / 0.2f;
        float cfy = (y - (-51.2f)) / 0.2f;
        bool m = (cfx >= 0.f) && (cfx < 512.f) && (cfy >= 0.f) && (cfy < 512.f);
        int cx = (int)floorf(cfx);
        int cy = (int)floorf(cfy);
        int b = (int)bi;
        int seg = m ? (b * NV_ + cy * GRID_ + cx) : (BATCH_ * NV_);

        float cv = fmaxf(cnt[seg], 1.0f);
        float mx = sxp[seg] / cv;
        float my = syp[seg] / cv;
        float mz = szp[seg] / cv;

        // 9 features (k=0..8), k=9..11 are zero padding
        float f0 = x, f1 = y, f2 = z, f3 = r;
        float f4 = x - mx, f5 = y - my, f6 = z - mz;
        float f7 = x - ((float)cx * 0.2f + 0.1f + (-51.2f));
        float f8 = y - ((float)cy * 0.2f + 0.1f + (-51.2f));

        // A fragments: lane holds row M=col, K = 4t + 2*half + {0,1}
        v2f a0 = (v2f){ half ? f2 : f0, half ? f3 : f1 };           // k 0..3
        v2f a1 = (v2f){ half ? f6 : f4, half ? f7 : f5 };           // k 4..7
        v2f a2 = (v2f){ half ? 0.f : f8, 0.f };                     // k 8..11 (padded)

        // Output base offset for this point: b*64*NV + cell, or -1 if masked/out-of-range
        int obase = (m && inb) ? (b * (COUT_ * NV_) + cy * GRID_ + cx) : -1;
        // Gather per-row (M) offsets across lanes: lane needs rows M = rr + 8*half
        int ob[8];
#pragma unroll
        for (int rr = 0; rr < 8; ++rr)
            ob[rr] = __shfl(obase, rr + 8 * half, 32);

#pragma unroll
        for (int nn = 0; nn < 4; ++nn) {
            v8f acc = (v8f){0.f, 0.f, 0.f, 0.f, 0.f, 0.f, 0.f, 0.f};
            acc = __builtin_amdgcn_wmma_f32_16x16x4_f32(false, a0, false, bf[nn][0], (short)0, acc, false, false);
            acc = __builtin_amdgcn_wmma_f32_16x16x4_f32(false, a1, false, bf[nn][1], (short)0, acc, false, false);
            acc = __builtin_amdgcn_wmma_f32_16x16x4_f32(false, a2, false, bf[nn][2], (short)0, acc, false, false);
            int coff = (16 * nn + col) * NV_;
            float s = scl[nn], bb = bia[nn];
#pragma unroll
            for (int rr = 0; rr < 8; ++rr) {
                float v = fmaxf(acc[rr] * s + bb, 0.f);   // BN + ReLU
                if (ob[rr] >= 0) {
                    // non-negative floats: uint ordering == float ordering; canvas init'd to 0
                    atomicMax((unsigned int*)(out + (size_t)ob[rr] + (size_t)coff),
                              __float_as_uint(v));
                }
            }
        }
    }
}

extern "C" void kernel_launch(void* const* d_in, const int* in_sizes, int n_in,
                              void* d_out, int out_size, void* d_ws, size_t ws_size,
                              hipStream_t stream) {
    const float* pts   = (const float*)d_in[0];
    const float* Wm    = (const float*)d_in[1];
    const float* gamma = (const float*)d_in[2];
    const float* beta  = (const float*)d_in[3];
    const float* rmean = (const float*)d_in[4];
    const float* rvar  = (const float*)d_in[5];
    float* out = (float*)d_out;
    float* ws  = (float*)d_ws;

    int n = in_sizes[0] / 5;

    long ws_elems = 4L * NSEG_;               // count + sum x/y/z
    zero_f<<<2048, 256, 0, stream>>>(ws, ws_elems);
    zero_f<<<8192, 256, 0, stream>>>(out, (long)out_size);

    pillar_stats<<<(n + 255) / 256, 256, 0, stream>>>(pts, ws, n);

    int ngroups = (n + 15) / 16;
    pillar_wmma<<<512, 256, 0, stream>>>(pts, Wm, gamma, beta, rmean, rvar, ws, out, n, ngroups);
}